// NNUE_16990890623528
// MI455X (gfx1250) — compile-verified
//
#include <hip/hip_runtime.h>

typedef __attribute__((ext_vector_type(2))) float v2f;
typedef __attribute__((ext_vector_type(8))) float v8f;

#define NUM_FEATURES 81920
#define BATCH        1024
#define CHUNK        2048            // K-chunk staged in LDS per block
#define SUBK         512             // K-range per wave (4 waves per matrix)
#define ROWTILES     (BATCH / 16)    // 64
#define KCHUNKS      (NUM_FEATURES / CHUNK) // 40
#define UF           8               // loads in flight per wave

// ---------------- Phase 0: zero the workspace accumulators -----------------
__global__ void nnue_zero_ws(float* __restrict__ p, int n) {
    int i = blockIdx.x * 256 + threadIdx.x;
    if (i < n) p[i] = 0.0f;
}

// ---------------- Phase 1: L0 GEMV via fp32 WMMA ---------------------------
// Each block: rowTile (16 batch rows) x kChunk (2048 K).
// Waves 0-3: white features, waves 4-7: black features; each wave covers 512 K.
__global__ __launch_bounds__(256) void nnue_l0_wmma(
    const float* __restrict__ whiteF,
    const float* __restrict__ blackF,
    const float* __restrict__ W0,
    float* __restrict__ accW,
    float* __restrict__ accB)
{
    __shared__ float w0s[5 * CHUNK];   // rows 0..3 = W0 chunk, row 4 = zeros

    const int rowTile = blockIdx.x % ROWTILES;
    const int kc      = blockIdx.x / ROWTILES;
    const int kcBase  = kc * CHUNK;

    // Cooperative stage of the W0 chunk (plus a zero row for B columns 4..15).
    for (int i = threadIdx.x; i < 5 * CHUNK; i += 256) {
        const int r = i / CHUNK;
        const int c = i % CHUNK;
        w0s[i] = (r < 4) ? W0[(size_t)r * NUM_FEATURES + kcBase + c] : 0.0f;
    }
    __syncthreads();

    const int wid   = threadIdx.x >> 5;
    const int lane  = threadIdx.x & 31;
    const int n     = lane & 15;     // A: row M; B: column N; C: column N
    const int hi    = lane >> 4;     // K half (A/B) and M half (C)
    const int koff  = hi * 2;        // low lanes K={0,1}, high lanes K={2,3}
    const int rowBase = rowTile * 16;
    const int mat   = wid >> 2;      // 0 = white, 1 = black
    const int sub   = wid & 3;       // K sub-range within the chunk

    const float* __restrict__ feat = mat ? blackF : whiteF;
    float*       __restrict__ acc  = mat ? accB  : accW;

    const int bRow = (n < 4) ? n : 4;              // row 4 is zero-filled
    // Hoisted base pointers so the inner loop has a compile-time trip count.
    const float* __restrict__ fptr =
        feat + (size_t)(rowBase + n) * NUM_FEATURES + kcBase + sub * SUBK + koff;
    const float* __restrict__ lptr = &w0s[bRow * CHUNK + sub * SUBK + koff];

    v8f c = {};
    // Batched pipeline: UF loads (global + LDS) issued together, then UF WMMAs.
    #pragma unroll
    for (int kk = 0; kk < SUBK; kk += 4 * UF) {
        v2f a[UF], b[UF];
        #pragma unroll
        for (int u = 0; u < UF; ++u) {
            a[u] = *(const v2f*)(fptr + kk + 4 * u);   // A: global_load_b64
            b[u] = *(const v2f*)(lptr + kk + 4 * u);   // B: ds_load_b64
        }
        #pragma unroll
        for (int u = 0; u < UF; ++u) {
            c = __builtin_amdgcn_wmma_f32_16x16x4_f32(
                    /*neg_a=*/false, a[u], /*neg_b=*/false, b[u],
                    /*c_mod=*/(short)0, c, /*reuse_a=*/false, /*reuse_b=*/false);
        }
    }

    // C layout: VGPR r -> M = r (lanes 0-15) / r+8 (lanes 16-31), N = lane&15.
    if (n < 4) {
        #pragma unroll
        for (int r = 0; r < 8; ++r) {
            atomicAdd(&acc[(size_t)(rowBase + r + hi * 8) * 4 + n], c[r]);
        }
    }
}

// ---------------- Phase 2: tiny MLP head + loss ----------------------------
__global__ void nnue_head(
    const float* __restrict__ accW, const float* __restrict__ accB,
    const float* __restrict__ turn, const float* __restrict__ score,
    const float* __restrict__ result,
    const float* __restrict__ b0,
    const float* __restrict__ W1, const float* __restrict__ b1,
    const float* __restrict__ W2, const float* __restrict__ b2,
    float* __restrict__ out)
{
    const int i = blockIdx.x * 256 + threadIdx.x;
    if (i >= BATCH) return;

    float w[4], bk[4];
    #pragma unroll
    for (int m = 0; m < 4; ++m) {
        w[m]  = accW[i * 4 + m] + b0[m];
        bk[m] = accB[i * 4 + m] + b0[m];
    }

    const float t = turn[i];
    float l1[8];
    #pragma unroll
    for (int j = 0; j < 8; ++j) {
        const float cw = (j < 4) ? w[j]  : bk[j - 4];
        const float cb = (j < 4) ? bk[j] : w[j - 4];
        const float v  = t * cw + (1.0f - t) * cb;
        l1[j] = fminf(fmaxf(v, 0.0f), 1.0f);
    }

    float l2[8];
    #pragma unroll
    for (int j = 0; j < 8; ++j) {
        float s = b1[j];
        #pragma unroll
        for (int k = 0; k < 8; ++k) s += W1[j * 8 + k] * l1[k];
        l2[j] = fminf(fmaxf(s, 0.0f), 1.0f);
    }

    float mr = b2[0];
    #pragma unroll
    for (int j = 0; j < 8; ++j) mr += W2[j] * l2[j];

    const float SCALING = 400.0f;
    const float LAMBDA  = 1.0f;
    const float wm = 1.0f / (1.0f + __expf(-mr / SCALING));
    const float wt = 1.0f / (1.0f + __expf(-score[i] / SCALING));
    const float le = (wm - wt) * (wm - wt);
    const float lr = (wm - result[i]) * (wm - result[i]);
    out[i] = LAMBDA * le + (1.0f - LAMBDA) * lr;
}

// ---------------------------------------------------------------------------
extern "C" void kernel_launch(void* const* d_in, const int* in_sizes, int n_in,
                              void* d_out, int out_size, void* d_ws, size_t ws_size,
                              hipStream_t stream) {
    const float* whiteF = (const float*)d_in[0];
    const float* blackF = (const float*)d_in[1];
    const float* turn   = (const float*)d_in[2];
    const float* score  = (const float*)d_in[3];
    const float* result = (const float*)d_in[4];
    const float* W0     = (const float*)d_in[5];
    const float* b0     = (const float*)d_in[6];
    const float* W1     = (const float*)d_in[7];
    const float* b1     = (const float*)d_in[8];
    const float* W2     = (const float*)d_in[9];
    const float* b2     = (const float*)d_in[10];

    float* accW = (float*)d_ws;           // [1024 x 4]
    float* accB = accW + BATCH * 4;       // [1024 x 4]

    nnue_zero_ws<<<(BATCH * 8 + 255) / 256, 256, 0, stream>>>(accW, BATCH * 8);

    nnue_l0_wmma<<<ROWTILES * KCHUNKS, 256, 0, stream>>>(
        whiteF, blackF, W0, accW, accB);

    nnue_head<<<(BATCH + 255) / 256, 256, 0, stream>>>(
        accW, accB, turn, score, result, b0, W1, b1, W2, b2, (float*)d_out);
}